// SplatModule_40020505264284
// MI455X (gfx1250) — compile-verified
//
#include <hip/hip_runtime.h>

// ---------------------------------------------------------------------------
// BEV splat (lift-splat pooling), MI455X / gfx1250.
//   coords_world:    [B,N,D,H,W,3]  f32
//   lifted_features: [B,N,D,C,H,W]  f32
//   out:             [B,C,200,200]  f32
// Strategy: per-(b,n,d)-slice blocks; coords staged to LDS via gfx1250 async
// to-LDS (ASYNCcnt path); bins computed once into LDS; features streamed with
// non-temporal float4 loads; scatter via global_atomic_add_f32 into a
// channel-contiguous [B, bin, C] accumulator (41 MB -> L2-resident on the
// 192 MB L2); tiled LDS transpose to the [B,C,H,W] output layout.
// ---------------------------------------------------------------------------

typedef float v4f __attribute__((ext_vector_type(4)));
typedef int   v4i __attribute__((ext_vector_type(4)));

constexpr int kB = 4, kN = 6, kD = 59, kC = 64, kH = 16, kW = 44;
constexpr int kHW = kH * kW;                 // 704 points per slice
constexpr int kSlices = kB * kN * kD;        // 1416
constexpr int kSlicesPerBatch = kN * kD;     // 354
constexpr int kBev = 200;
constexpr int kBins = kBev * kBev;           // 40000
constexpr long kOutPerBatch = (long)kBins * kC;   // 2,560,000
constexpr long kOutTotal = (long)kB * kOutPerBatch; // 10,240,000 floats

#if defined(__gfx1250__) && __has_builtin(__builtin_amdgcn_global_load_async_to_lds_b128)
#define BEV_ASYNC_LDS 1
#else
#define BEV_ASYNC_LDS 0
#endif

#if BEV_ASYNC_LDS
typedef __attribute__((address_space(1))) char  gchar_t;
typedef __attribute__((address_space(3))) char  lchar_t;
typedef __attribute__((address_space(1))) v4i   gv4i_t;
typedef __attribute__((address_space(3))) v4i   lv4i_t;
#endif

__device__ __forceinline__ void wait_asynccnt0() {
#if defined(__gfx1250__)
#if __has_builtin(__builtin_amdgcn_s_wait_asynccnt)
  __builtin_amdgcn_s_wait_asynccnt(0);
#else
  asm volatile("s_wait_asynccnt 0" ::: "memory");
#endif
#endif
}

// Shared phase A: stage coords slice into LDS (async), bin all 704 points.
__device__ __forceinline__ void bin_slice(const float* __restrict__ coords,
                                          int slice, int t,
                                          float* s_coords, int* s_bin) {
  const float* cbase = coords + (size_t)slice * kHW * 3;
#if BEV_ASYNC_LDS
  {
    gchar_t* gp = (gchar_t*)cbase;
    lchar_t* lp = (lchar_t*)s_coords;
    for (int off = t * 16; off < kHW * 3 * 4; off += 256 * 16) {
      __builtin_amdgcn_global_load_async_to_lds_b128(
          (gv4i_t*)(gp + off), (lv4i_t*)(lp + off), 0, 0);
    }
    wait_asynccnt0();
  }
#else
  for (int i = t; i < kHW * 3; i += 256) s_coords[i] = cbase[i];
#endif
  __syncthreads();

  for (int p = t; p < kHW; p += 256) {
    const float x = s_coords[p * 3 + 0];
    const float y = s_coords[p * 3 + 1];
    const float fx = (x + 50.0f) / 100.0f * 200.0f;   // matches reference math
    const float fy = (y + 50.0f) / 100.0f * 200.0f;
    const int xi = (int)fx;   // C cast == trunc toward zero (jnp.trunc)
    const int yi = (int)fy;
    const bool valid = (xi >= 0) & (xi < kBev) & (yi >= 0) & (yi < kBev);
    s_bin[p] = valid ? (yi * kBev + xi) : -1;
  }
  __syncthreads();
}

// Main path: accumulate into workspace with channel-contiguous [B, bin, C].
__global__ __launch_bounds__(256) void bev_scatter_ws(const float* __restrict__ coords,
                                                      const float* __restrict__ feats,
                                                      float* __restrict__ dst) {
  __shared__ alignas(16) float s_coords[kHW * 3];   // 8448 B
  __shared__ int s_bin[kHW];                        // 2816 B

  const int slice = blockIdx.x;
  const int b     = slice / kSlicesPerBatch;
  const int t     = threadIdx.x;

  bin_slice(coords, slice, t, s_coords, s_bin);

  // One lane owns 4 consecutive points (float4 along contiguous W axis),
  // loops all 64 channels: coalesced 512 B/wave non-temporal reads; each
  // in-range point does 64 atomics into a contiguous 256 B region.
  const float* fbase = feats + (size_t)slice * kC * kHW;
  if (t < kHW / 4) {
    const int p0 = t << 2;
    const int b0 = s_bin[p0 + 0];
    const int b1 = s_bin[p0 + 1];
    const int b2 = s_bin[p0 + 2];
    const int b3 = s_bin[p0 + 3];
    if ((b0 >= 0) | (b1 >= 0) | (b2 >= 0) | (b3 >= 0)) {
      const float* fp = fbase + p0;
      float* wsb = dst + (size_t)b * kOutPerBatch;
#pragma unroll 4
      for (int c = 0; c < kC; ++c) {
        const v4f v = __builtin_nontemporal_load((const v4f*)(fp + c * kHW));
        if (b0 >= 0) atomicAdd(wsb + (((size_t)b0) << 6) + c, v.x);
        if (b1 >= 0) atomicAdd(wsb + (((size_t)b1) << 6) + c, v.y);
        if (b2 >= 0) atomicAdd(wsb + (((size_t)b2) << 6) + c, v.z);
        if (b3 >= 0) atomicAdd(wsb + (((size_t)b3) << 6) + c, v.w);
      }
    }
  }
}

// Fallback path: accumulate directly into d_out ([B, C, bin] layout).
__global__ __launch_bounds__(256) void bev_scatter_direct(const float* __restrict__ coords,
                                                          const float* __restrict__ feats,
                                                          float* __restrict__ dst) {
  __shared__ alignas(16) float s_coords[kHW * 3];
  __shared__ int s_bin[kHW];

  const int slice = blockIdx.x;
  const int b     = slice / kSlicesPerBatch;
  const int t     = threadIdx.x;

  bin_slice(coords, slice, t, s_coords, s_bin);

  const float* fbase = feats + (size_t)slice * kC * kHW;
  if (t < kHW / 4) {
    const int p0 = t << 2;
    const int b0 = s_bin[p0 + 0];
    const int b1 = s_bin[p0 + 1];
    const int b2 = s_bin[p0 + 2];
    const int b3 = s_bin[p0 + 3];
    if ((b0 >= 0) | (b1 >= 0) | (b2 >= 0) | (b3 >= 0)) {
      const float* fp = fbase + p0;
      float* ob = dst + (size_t)b * kOutPerBatch;   // [C, bins] within batch
#pragma unroll 4
      for (int c = 0; c < kC; ++c) {
        const v4f v = __builtin_nontemporal_load((const v4f*)(fp + c * kHW));
        float* cp = ob + (size_t)c * kBins;
        if (b0 >= 0) atomicAdd(cp + b0, v.x);
        if (b1 >= 0) atomicAdd(cp + b1, v.y);
        if (b2 >= 0) atomicAdd(cp + b2, v.z);
        if (b3 >= 0) atomicAdd(cp + b3, v.w);
      }
    }
  }
}

// [B, bin, C] -> [B, C, bin] tiled transpose; 64x64 tiles, LDS 64x65 padding.
__global__ __launch_bounds__(256) void bev_transpose(const float* __restrict__ ws,
                                                     float* __restrict__ out) {
  __shared__ float tile[64][65];
  const int bidx = blockIdx.x;             // b * 625 + tileIdx (40000/64 = 625)
  const int b    = bidx / 625;
  const int hw0  = (bidx % 625) * 64;
  const int t    = threadIdx.x;

  const float* src = ws + ((size_t)b * kBins + hw0) * kC;
#pragma unroll
  for (int k = 0; k < 16; ++k) {
    const int idx = k * 256 + t;
    const int r = idx >> 6;        // local hw
    const int c = idx & 63;        // channel
    tile[r][c] = src[idx];         // coalesced along c
  }
  __syncthreads();

  float* dstb = out + (size_t)b * kOutPerBatch + hw0;
#pragma unroll
  for (int k = 0; k < 16; ++k) {
    const int idx = k * 256 + t;
    const int c = idx >> 6;        // channel
    const int w = idx & 63;        // local hw
    dstb[(size_t)c * kBins + w] = tile[w][c];   // coalesced along hw
  }
}

__global__ __launch_bounds__(256) void bev_zero(float* __restrict__ p, long n) {
  const long i = ((long)blockIdx.x * 256 + threadIdx.x) * 4;
  if (i + 3 < n) {
    const v4f z = {0.0f, 0.0f, 0.0f, 0.0f};
    *(v4f*)(p + i) = z;
  }
}

extern "C" void kernel_launch(void* const* d_in, const int* in_sizes, int n_in,
                              void* d_out, int out_size, void* d_ws, size_t ws_size,
                              hipStream_t stream) {
  const float* coords = (const float*)d_in[0];
  const float* feats  = (const float*)d_in[1];
  float* out = (float*)d_out;

  const size_t need = (size_t)kOutTotal * sizeof(float);
  const int zeroBlocks = (int)((kOutTotal / 4 + 255) / 256);

  if (ws_size >= need) {
    float* ws = (float*)d_ws;
    bev_zero<<<zeroBlocks, 256, 0, stream>>>(ws, kOutTotal);
    bev_scatter_ws<<<kSlices, 256, 0, stream>>>(coords, feats, ws);
    bev_transpose<<<kB * 625, 256, 0, stream>>>(ws, out);
  } else {
    // Workspace too small: accumulate straight into the output layout.
    bev_zero<<<zeroBlocks, 256, 0, stream>>>(out, kOutTotal);
    bev_scatter_direct<<<kSlices, 256, 0, stream>>>(coords, feats, out);
  }
}